// IWNewMaxSquareloss_20512763806266
// MI455X (gfx1250) — compile-verified
//
#include <hip/hip_runtime.h>

// IWNewMaxSquareloss for MI455X (gfx1250).
// Single pass over the 319MB input via TDM (tensor_load_to_lds) double
// buffering into LDS; accumulates per-(n,c) sum(probs^2) and argmax histogram;
// tiny finalize kernel computes class weights and the scalar loss.
// All TDM issue control flow is scalar (readfirstlane-derived) because TDM
// ignores EXEC; the last tile is peeled so no conditional issue exists.

#define NIMG 8
#define CCH 19
#define HWPIX (512 * 1024)          // 524288 pixels per (n, c) plane
#define TILE_PIX 64                 // pixels per TDM tile (row = 256B)
#define WPB 4                       // waves per block
#define BLOCK_T (WPB * 32)
#define BPI 256                     // blocks per image -> 1024 waves/image
#define TILES (HWPIX / TILE_PIX)    // 8192 tiles per image
#define TILE_FLOATS (CCH * TILE_PIX)

static_assert(BPI * WPB <= TILES, "every wave must own at least one tile");

typedef unsigned int u32;
typedef unsigned long long u64;
typedef u32 u32x4 __attribute__((ext_vector_type(4)));
typedef int i32x4 __attribute__((ext_vector_type(4)));
typedef int i32x8 __attribute__((ext_vector_type(8)));

// Issue one TDM load of a (TILE_PIX x CCH) 2D tile (row stride = HWPIX elems)
// from global into LDS at byte offset ldsByte. One TENSORcnt token per call.
// cpol=1 -> TH=NT (stream-once data, keep L2 clean).
__device__ __forceinline__ void tdm_issue(const float* gsrc, u32 ldsByte) {
  u64 ga = (u64)(const void*)gsrc;
  u32x4 g0;
  g0.x = 1u;                                              // count=1, user desc
  g0.y = ldsByte;                                         // lds_addr (bytes)
  g0.z = (u32)ga;                                         // global_addr[31:0]
  g0.w = (u32)((ga >> 32) & 0x1FFFFFFull) | (2u << 30);   // addr[56:32] | type=2
  i32x8 g1;
  g1[0] = (int)(2u << 16);                                // data_size=4B, mask=0
  g1[1] = (int)((HWPIX & 0xFFFF) << 16);                  // tensor_dim0[15:0]
  g1[2] = (int)(((HWPIX >> 16) & 0xFFFF) | (CCH << 16));  // dim0[31:16] | dim1 lo
  g1[3] = (int)(TILE_PIX << 16);                          // dim1 hi(=0) | tile_dim0
  g1[4] = (int)CCH;                                       // tile_dim1 | tile_dim2=0
  g1[5] = (int)(HWPIX & 0xFFFFFFFFu);                     // dim0_stride[31:0]
  g1[6] = 0;                                              // stride hi, dim1_stride lo
  g1[7] = 0;
  i32x4 gz = {0, 0, 0, 0};
#if defined(__clang_major__) && __clang_major__ >= 23
  i32x8 gz8 = {0, 0, 0, 0, 0, 0, 0, 0};
  __builtin_amdgcn_tensor_load_to_lds(g0, g1, gz, gz, gz8, 1);
#else
  __builtin_amdgcn_tensor_load_to_lds(g0, g1, gz, gz, 1);
#endif
}

// Consume one 19x64 tile from LDS: per-lane 2 pixels, softmax, per-class
// sum(p^2) into acc[], argmax histogram into LDS hist[].
__device__ __forceinline__ void consume_tile(const float* tp, int lane,
                                             float* acc, u32* hist) {
  float2 vals[CCH];
  float m0 = -3.402823466e38f, m1 = -3.402823466e38f;
  int i0 = 0, i1 = 0;
#pragma unroll
  for (int c = 0; c < CCH; ++c) {
    float2 v = *(const float2*)(tp + c * TILE_PIX + 2 * lane);  // ds_load_b64
    vals[c] = v;
    if (v.x > m0) { m0 = v.x; i0 = c; }   // first-max semantics (strict >)
    if (v.y > m1) { m1 = v.y; i1 = c; }
  }
  const float L2E = 1.4426950408889634f;
  float s0 = 0.f, s1 = 0.f;
#pragma unroll
  for (int c = 0; c < CCH; ++c) {
    float e0 = __builtin_exp2f((vals[c].x - m0) * L2E);
    float e1 = __builtin_exp2f((vals[c].y - m1) * L2E);
    vals[c].x = e0; vals[c].y = e1;
    s0 += e0; s1 += e1;
  }
  float q0 = __builtin_amdgcn_rcpf(s0); q0 *= q0;  // 1/denom^2
  float q1 = __builtin_amdgcn_rcpf(s1); q1 *= q1;
#pragma unroll
  for (int c = 0; c < CCH; ++c) {
    acc[c] = fmaf(vals[c].x * vals[c].x, q0, acc[c]);  // += p^2
    acc[c] = fmaf(vals[c].y * vals[c].y, q1, acc[c]);
  }
  atomicAdd(&hist[i0], 1u);   // ds_add_u32
  atomicAdd(&hist[i1], 1u);
}

__global__ void __launch_bounds__(BLOCK_T)
iw_main(const float* __restrict__ in, float* __restrict__ Sg, u32* __restrict__ Hg) {
  __shared__ float tiles[WPB * 2 * TILE_FLOATS];  // per-wave double buffers
  __shared__ float sacc[CCH];
  __shared__ u32 hist[CCH];

  const int tid = threadIdx.x;
  const int lane = tid & 31;
  // scalar (SGPR-uniform) wave id -> all TDM control flow is scalar branches
  const int wave = __builtin_amdgcn_readfirstlane(tid) >> 5;
  const int n = blockIdx.y;

  if (tid < CCH) { sacc[tid] = 0.f; hist[tid] = 0u; }
  __syncthreads();

  const float* imgBase = in + (size_t)n * CCH * HWPIX;
  float* wbuf0 = &tiles[(wave * 2 + 0) * TILE_FLOATS];
  float* wbuf1 = &tiles[(wave * 2 + 1) * TILE_FLOATS];
  const u32 ldsA = (u32)(size_t)(void*)wbuf0;  // flat LDS addr low32 == LDS offset
  const u32 ldsB = (u32)(size_t)(void*)wbuf1;

  const int step = (int)gridDim.x * WPB;          // uniform
  int t = (int)blockIdx.x * WPB + wave;           // uniform (scalar wave id)

  float acc[CCH];
#pragma unroll
  for (int c = 0; c < CCH; ++c) acc[c] = 0.f;

  // prologue: every wave owns at least one tile (static_assert above)
  tdm_issue(imgBase + (size_t)t * TILE_PIX, ldsA);
  int cur = 0;

  // steady state: issue tile t+step unconditionally (guard guarantees validity)
  for (; t + step < TILES; t += step) {
    // drain ds_loads from the buffer about to be overwritten by the TDM engine
    asm volatile("s_wait_dscnt 0" ::: "memory");
    tdm_issue(imgBase + (size_t)(t + step) * TILE_PIX, cur ? ldsA : ldsB);
    __builtin_amdgcn_s_wait_tensorcnt(1);   // tile t has landed (in-order)
    asm volatile("" ::: "memory");
    consume_tile(cur ? wbuf1 : wbuf0, lane, acc, hist);
    cur ^= 1;
  }

  // epilogue: last tile, no further issue
  __builtin_amdgcn_s_wait_tensorcnt(0);
  asm volatile("" ::: "memory");
  consume_tile(cur ? wbuf1 : wbuf0, lane, acc, hist);

#pragma unroll
  for (int c = 0; c < CCH; ++c) atomicAdd(&sacc[c], acc[c]);  // ds_add_f32
  __syncthreads();

  if (tid < CCH) {
    atomicAdd(&Sg[n * CCH + tid], sacc[tid]);   // global_atomic_add_f32
    atomicAdd(&Hg[n * CCH + tid], hist[tid]);   // global_atomic_add_u32
  }
}

__global__ void __launch_bounds__(256) iw_init_ws(float* Sg, u32* Hg, float* out) {
  int t = blockIdx.x * blockDim.x + threadIdx.x;
  if (t < NIMG * CCH) { Sg[t] = 0.f; Hg[t] = 0u; }
  if (t == 0) out[0] = 0.f;
}

__global__ void __launch_bounds__(32)
iw_finalize(const float* __restrict__ Sg, const u32* __restrict__ Hg, float* out) {
  if (threadIdx.x == 0 && blockIdx.x == 0) {
    float total = 0.f;
#pragma clang loop unroll(disable)
    for (int n = 0; n < NIMG; ++n) {
      float h[CCH];
      float hs = 0.f;
#pragma clang loop unroll(disable)
      for (int c = 0; c < CCH; ++c) {
        float v = (float)Hg[n * CCH + c];
        if (v == 0.f) v = 1.f;        // hist[hist==0] = 1 before row sum
        h[c] = v;
        hs += v;
      }
#pragma clang loop unroll(disable)
      for (int c = 0; c < CCH; ++c) {
        float w = (c < 11) ? 1.f : __powf(hs / h[c], 0.2f);  // OLD_CL=11, RATIO=0.2
        total += Sg[n * CCH + c] * w;
      }
    }
    out[0] = -total / (float)(NIMG * CCH) / (float)HWPIX;
  }
}

extern "C" void kernel_launch(void* const* d_in, const int* in_sizes, int n_in,
                              void* d_out, int out_size, void* d_ws, size_t ws_size,
                              hipStream_t stream) {
  (void)in_sizes; (void)n_in; (void)out_size; (void)ws_size;
  const float* in = (const float*)d_in[0];
  float* out = (float*)d_out;
  float* Sg = (float*)d_ws;                                    // 152 floats
  u32* Hg = (u32*)((char*)d_ws + NIMG * CCH * sizeof(float));  // 152 u32

  hipLaunchKernelGGL(iw_init_ws, dim3(1), dim3(256), 0, stream, Sg, Hg, out);
  hipLaunchKernelGGL(iw_main, dim3(BPI, NIMG), dim3(BLOCK_T), 0, stream, in, Sg, Hg);
  hipLaunchKernelGGL(iw_finalize, dim3(1), dim3(32), 0, stream, Sg, Hg, out);
}